// DGMC_45037027066323
// MI455X (gfx1250) — compile-verified
//
#include <hip/hip_runtime.h>
#include <hip/hip_bf16.h>

typedef __attribute__((ext_vector_type(2))) float v2f;
typedef __attribute__((ext_vector_type(4))) float v4f;
typedef __attribute__((ext_vector_type(8))) float v8f;

namespace {
constexpr int kNs       = 10000;
constexpr int kNt       = 10000;
constexpr int kC        = 256;
constexpr int kK        = 10;     // top-k
constexpr int kColTiles = kNt / 16;   // 625
constexpr int kRowTiles = kNs / 16;   // 625
constexpr int kWaves    = 4;
constexpr int kThreads  = kWaves * 32;

// LDS staging pitches (floats). Row stride mod 64 banks == 4 -> the per-lane
// ds_load_b64 pattern (lane r<16 -> banks {4r,4r+1}, lane r+16 -> {4r+2,4r+3})
// touches all 64 banks with no conflict.
constexpr int kApitch   = 260;    // full K=256 rows of A (+4 pad)
constexpr int kBpitch   = 132;    // K staged in two 128-wide chunks (+4 pad)
constexpr int kCpitch   = 17;     // 16x16 S-tile scratch (+1 pad)
constexpr int kAfloats  = 16 * kApitch;           // 4160
constexpr int kBfloats  = 16 * kBpitch;           // 2112
constexpr int kCfloats  = 16 * kCpitch;           // 272
constexpr int kWaveFl   = kBfloats + kCfloats;    // 2384
constexpr int kSmemFl   = kAfloats + kWaves * kWaveFl;  // 13696 floats = 54.8 KB
constexpr int kCandPerRow = kWaves * 2 * kK;      // 80 merge candidates per row
} // namespace

__global__ __launch_bounds__(kThreads)
void dgmc_topk_corr(const float* __restrict__ ehs, const float* __restrict__ eht,
                    const float* __restrict__ rhs, const float* __restrict__ rht,
                    float* __restrict__ out)
{
    __shared__ __align__(16) float smem[kSmemFl];

    const int tid  = threadIdx.x;
    const int lane = tid & 31;
    const int wave = tid >> 5;
    const int arow = lane & 15;   // M (for A) / N (for B) / owned S row
    const int hi   = lane >> 4;   // K-subgroup select, and owned column half
    const int rowBase = blockIdx.x * 16;
    const int pair    = blockIdx.y;          // 0 = e-tensors, 1 = r-tensors

    const float* hs = pair ? rhs : ehs;
    const float* ht = pair ? rht : eht;
    float* valOut = out + pair * (kNs * kK);
    float* idxOut = out + 2 * (kNs * kK) + pair * (kNs * kK);

    float* ldsA = smem;
    float* ldsB = smem + kAfloats + wave * kWaveFl;
    float* ldsC = ldsB + kBfloats;

    // ---- Stage A = hs[rowBase : rowBase+16, :] (16 KB, contiguous) ----
    {
        const v4f* src = (const v4f*)(hs + (size_t)rowBase * kC);
        #pragma unroll
        for (int p = 0; p < (16 * kC / 4) / kThreads; ++p) {   // 8 iters
            int i = p * kThreads + tid;
            v4f v = src[i];
            *(v4f*)&ldsA[(i >> 6) * kApitch + ((i & 63) << 2)] = v;
        }
    }
    __syncthreads();

    // ---- Per-lane running top-10 for (row = arow, columns ≡ half hi) ----
    float lv[kK];
    int   li[kK];
    #pragma unroll
    for (int t = 0; t < kK; ++t) { lv[t] = -__builtin_inff(); li[t] = 0; }

    const int aBase = arow * kApitch + 2 * hi;
    const int bBase = arow * kBpitch + 2 * hi;

    for (int it = wave; it < kColTiles; it += kWaves) {
        const int n0 = it * 16;
        v8f c = {0.f, 0.f, 0.f, 0.f, 0.f, 0.f, 0.f, 0.f};

        #pragma unroll
        for (int kh = 0; kh < 2; ++kh) {
            // Stage B chunk: ht rows [n0,n0+16), cols [kh*128, kh*128+128).
            // One coalesced 512B b128 load per row per wave.
            #pragma unroll 4
            for (int r = 0; r < 16; ++r) {
                v4f v = *(const v4f*)(ht + ((size_t)(n0 + r)) * kC + kh * 128 + (lane << 2));
                *(v4f*)&ldsB[r * kBpitch + (lane << 2)] = v;
            }
            // Cross-lane LDS RAW within the wave: wait for DS stores.
            __asm__ __volatile__("s_wait_dscnt 0x0" ::: "memory");

            #pragma unroll 8
            for (int kk = 0; kk < 128; kk += 4) {
                v2f a = *(const v2f*)&ldsA[aBase + kh * 128 + kk];
                v2f b = *(const v2f*)&ldsB[bBase + kk];
                // D = A(16x4 f32) * B(4x16 f32) + C, exact fp32 WMMA
                c = __builtin_amdgcn_wmma_f32_16x16x4_f32(
                        false, a, false, b, (short)0, c, false, false);
            }
        }

        // ---- Dump 16x16 S tile to LDS scratch (C layout: vgpr i holds
        //      (M = i + 8*hi, N = arow) for this lane) ----
        #pragma unroll
        for (int i = 0; i < 8; ++i)
            ldsC[(hi * 8 + i) * kCpitch + arow] = c[i];
        __asm__ __volatile__("s_wait_dscnt 0x0" ::: "memory");

        // ---- Streaming top-10 insertion: lane owns row `arow`, cols hi*8.. ----
        #pragma unroll
        for (int j = 0; j < 8; ++j) {
            float v  = ldsC[arow * kCpitch + hi * 8 + j];
            int   ix = n0 + hi * 8 + j;
            if (v > lv[kK - 1]) {
                #pragma unroll
                for (int t = 0; t < kK; ++t) {
                    if (v > lv[t]) {
                        float tv = lv[t]; int ti = li[t];
                        lv[t] = v; li[t] = ix;
                        v = tv; ix = ti;
                    }
                }
            }
        }
    }

    // ---- Merge: overlay candidate pool on the (dead) A region ----
    __syncthreads();
    float* mVal = smem;                                   // 16*80 floats
    int*   mIdx = (int*)(smem + 16 * kCandPerRow);        // 16*80 ints
    {
        const int slot = (wave * 2 + hi) * kK;
        #pragma unroll
        for (int t = 0; t < kK; ++t) {
            mVal[arow * kCandPerRow + slot + t] = lv[t];
            mIdx[arow * kCandPerRow + slot + t] = li[t];
        }
    }
    __syncthreads();

    if (tid < 16) {
        float fv[kK]; int fi[kK];
        #pragma unroll
        for (int t = 0; t < kK; ++t) { fv[t] = -__builtin_inff(); fi[t] = 0; }
        for (int j = 0; j < kCandPerRow; ++j) {
            float v  = mVal[tid * kCandPerRow + j];
            int   ix = mIdx[tid * kCandPerRow + j];
            if (v > fv[kK - 1]) {
                #pragma unroll
                for (int t = 0; t < kK; ++t) {
                    if (v > fv[t]) {
                        float tv = fv[t]; int ti = fi[t];
                        fv[t] = v; fi[t] = ix;
                        v = tv; ix = ti;
                    }
                }
            }
        }
        // softmax over the k=10 (descending) candidates
        float mx = fv[0];
        float ex[kK];
        float s = 0.f;
        #pragma unroll
        for (int t = 0; t < kK; ++t) { ex[t] = __expf(fv[t] - mx); s += ex[t]; }
        const float inv = 1.0f / s;
        const int row = rowBase + tid;
        #pragma unroll
        for (int t = 0; t < kK; ++t) {
            valOut[row * kK + t] = ex[t] * inv;
            idxOut[row * kK + t] = (float)fi[t];
        }
    }
}

extern "C" void kernel_launch(void* const* d_in, const int* in_sizes, int n_in,
                              void* d_out, int out_size, void* d_ws, size_t ws_size,
                              hipStream_t stream) {
    const float* ehs = (const float*)d_in[0];
    const float* eht = (const float*)d_in[1];
    const float* rhs = (const float*)d_in[2];
    const float* rht = (const float*)d_in[3];
    float* out = (float*)d_out;

    dim3 grid(kRowTiles, 2);   // 625 row-blocks x {e, r}
    dgmc_topk_corr<<<grid, kThreads, 0, stream>>>(ehs, eht, rhs, rht, out);
}